// _TripletMessage_16784732193362
// MI455X (gfx1250) — compile-verified
//
#include <hip/hip_runtime.h>
#include <stdint.h>

typedef __attribute__((ext_vector_type(2))) float v2f;
typedef __attribute__((ext_vector_type(8))) float v8f;

#define N_NODES   50000
#define N_EDGES   800000
#define CH        64       // node channels C
#define HEADS     3        // H
#define ECH       32       // edge channels
#define HC        (HEADS*CH)   // 192
#define NEG_SLOPE 0.2f

// ---------------------------------------------------------------------------
// fp32 WMMA GEMM: C[M,NCOLS] = A[M,KDIM] * B[KDIM,NCOLS] (+ bias)
// One wave computes one 16x16 tile via V_WMMA_F32_16X16X4_F32, K stepped by 4.
// Requires: M % 16 == 0 (true for all call sites: 50000, 800000).
// KDIM/NCOLS are compile-time so every load/store is base + immediate offset.
//
// fp32 WMMA operand layout (wave32):
//   A (16x4):  lane l, vgpr v -> A[M = l&15, K = 2*(l>>4) + v]   (v2f = b64 load)
//   B (4x16):  lane l, vgpr v -> B[K = 2*(l>>4) + v, N = l&15]
//   C/D:       lane l, vgpr v -> C[M = v + 8*(l>>4), N = l&15]
// ---------------------------------------------------------------------------
template<int KDIM, int NCOLS, bool HAS_BIAS>
__global__ __launch_bounds__(256) void wmma_gemm_f32(
    const float* __restrict__ A, const float* __restrict__ B,
    float* __restrict__ C, const float* __restrict__ bias, int M)
{
    const int lane = threadIdx.x & 31;
    const int wave = threadIdx.x >> 5;
    constexpr int tilesN = NCOLS / 16;
    const int tilesM = M >> 4;
    const int tile = blockIdx.x * (blockDim.x >> 5) + wave;
    if (tile >= tilesM * tilesN) return;      // wave-uniform: EXEC all-1 for WMMA

    const int tm = tile / tilesN;
    const int tn = tile - tm * tilesN;
    const int row0 = tm << 4;
    const int col0 = tn << 4;

    const int half = lane >> 4;               // 0 or 1
    const int l15  = lane & 15;
    const int koff = half << 1;               // 0 or 2

    const float* __restrict__ Ap = A + (size_t)(row0 + l15) * KDIM + koff;
    const float* __restrict__ Bp = B + (size_t)koff * NCOLS + (col0 + l15);

    v8f acc = {};
#pragma unroll
    for (int k0 = 0; k0 < KDIM; k0 += 4) {
        const v2f a = *(const v2f*)(Ap + k0);          // global_load_b64, imm offset
        v2f b;
        b.x = Bp[(size_t)(k0    ) * NCOLS];            // imm offsets (NCOLS constexpr)
        b.y = Bp[(size_t)(k0 + 1) * NCOLS];
        acc = __builtin_amdgcn_wmma_f32_16x16x4_f32(
            /*neg_a=*/false, a, /*neg_b=*/false, b,
            /*c_mod=*/(short)0, acc, /*reuse_a=*/false, /*reuse_b=*/false);
    }

    float* __restrict__ Cp = C + (size_t)(row0 + (half << 3)) * NCOLS + (col0 + l15);
    float bv = 0.0f;
    if constexpr (HAS_BIAS) bv = bias[col0 + l15];
#pragma unroll
    for (int v = 0; v < 8; ++v)
        Cp[(size_t)v * NCOLS] = acc[v] + bv;
}

// ---------------------------------------------------------------------------
// Per-call state init (harness does not re-poison between replays)
// ---------------------------------------------------------------------------
__global__ void init_state(float* __restrict__ aggr, float* __restrict__ segsum,
                           unsigned* __restrict__ segmax)
{
    const int i = blockIdx.x * blockDim.x + threadIdx.x;
    if (i < N_NODES * HC) aggr[i] = 0.0f;
    if (i < N_NODES * HEADS) { segsum[i] = 0.0f; segmax[i] = 0u; }
}

// monotonic float<->uint mapping for atomicMax on floats
__device__ __forceinline__ unsigned f32_ord_encode(float f) {
    unsigned u = __float_as_uint(f);
    return (u & 0x80000000u) ? ~u : (u | 0x80000000u);
}
__device__ __forceinline__ float f32_ord_decode(unsigned u) {
    return (u & 0x80000000u) ? __uint_as_float(u & 0x7fffffffu)
                             : __uint_as_float(~u);
}

// ---------------------------------------------------------------------------
// Pass A: one wave per edge. alpha[e,h] = leakyrelu( xi.w0 + e.w1 + xj.w2 ),
// encoded atomicMax into segmax[dst,h].
// ---------------------------------------------------------------------------
__global__ __launch_bounds__(256) void alpha_kernel(
    const float* __restrict__ xp, const float* __restrict__ ep,
    const long long* __restrict__ eidx, const float* __restrict__ Wt,
    float* __restrict__ alpha, unsigned* __restrict__ segmax)
{
    const int lane = threadIdx.x & 31;
    const int e = blockIdx.x * (blockDim.x >> 5) + (threadIdx.x >> 5);
    if (e >= N_EDGES) return;

    const int src = (int)eidx[e];
    const int dst = (int)eidx[N_EDGES + e];
    const float* __restrict__ xi = xp + (size_t)dst * HC;
    const float* __restrict__ xj = xp + (size_t)src * HC;
    const float* __restrict__ ee = ep + (size_t)e   * HC;

#pragma unroll
    for (int h = 0; h < HEADS; ++h) {
        const float* __restrict__ w = Wt + h * (3 * CH);  // [w_xi | w_e | w_xj]
        float p = 0.0f;
#pragma unroll
        for (int s = 0; s < CH / 32; ++s) {               // 2 channels / lane
            const int c = lane + s * 32;
            p += xi[h * CH + c] * w[c];
            p += ee[h * CH + c] * w[CH + c];
            p += xj[h * CH + c] * w[2 * CH + c];
        }
#pragma unroll
        for (int off = 16; off > 0; off >>= 1)            // wave32 reduction
            p += __shfl_xor(p, off, 32);
        if (lane == 0) {
            const float a = (p >= 0.0f) ? p : NEG_SLOPE * p;
            alpha[(size_t)e * HEADS + h] = a;
            atomicMax(&segmax[dst * HEADS + h], f32_ord_encode(a));
        }
    }
}

// ---------------------------------------------------------------------------
// Pass B: alpha <- exp(alpha - segmax[dst]); segsum[dst] += alpha  (f32 atomic)
// ---------------------------------------------------------------------------
__global__ __launch_bounds__(256) void softmax_expsum(
    float* __restrict__ alpha, const unsigned* __restrict__ segmax,
    const long long* __restrict__ eidx, float* __restrict__ segsum)
{
    const int i = blockIdx.x * blockDim.x + threadIdx.x;
    if (i >= N_EDGES * HEADS) return;
    const int e = i / HEADS;
    const int h = i - e * HEADS;
    const int dst = (int)eidx[N_EDGES + e];
    const float m = f32_ord_decode(segmax[dst * HEADS + h]);
    const float v = __expf(alpha[i] - m);
    alpha[i] = v;
    unsafeAtomicAdd(&segsum[dst * HEADS + h], v);
}

// ---------------------------------------------------------------------------
// Pass C: one wave per edge. msg = (alpha/segsum) * e_ij * x_j, scatter-add
// into aggr[dst] with native f32 global atomics (L2-resident).
// ---------------------------------------------------------------------------
__global__ __launch_bounds__(256) void scatter_msg(
    const float* __restrict__ xp, const float* __restrict__ ep,
    const long long* __restrict__ eidx, const float* __restrict__ alpha,
    const float* __restrict__ segsum, float* __restrict__ aggr)
{
    const int lane = threadIdx.x & 31;
    const int e = blockIdx.x * (blockDim.x >> 5) + (threadIdx.x >> 5);
    if (e >= N_EDGES) return;

    const int src = (int)eidx[e];
    const int dst = (int)eidx[N_EDGES + e];

    float coef[HEADS];
#pragma unroll
    for (int h = 0; h < HEADS; ++h)
        coef[h] = alpha[(size_t)e * HEADS + h] /
                  (segsum[dst * HEADS + h] + 1e-16f);

    const float* __restrict__ xj  = xp + (size_t)src * HC;
    const float* __restrict__ ee  = ep + (size_t)e   * HC;
    float* __restrict__ out = aggr + (size_t)dst * HC;

#pragma unroll
    for (int s = 0; s < HC / 32; ++s) {                    // 6 channels / lane
        const int c = lane + s * 32;
        const float m = coef[c >> 6] * ee[c] * xj[c];
        unsafeAtomicAdd(&out[c], m);
    }
}

// ---------------------------------------------------------------------------
extern "C" void kernel_launch(void* const* d_in, const int* in_sizes, int n_in,
                              void* d_out, int out_size, void* d_ws, size_t ws_size,
                              hipStream_t stream)
{
    const float*     x      = (const float*)d_in[0];      // [N, C]
    const long long* eidx   = (const long long*)d_in[1];  // [2, E] int64
    const float*     eattr  = (const float*)d_in[2];      // [E, EC]
    const float*     Wn     = (const float*)d_in[3];      // [C, H*C]
    const float*     We     = (const float*)d_in[4];      // [EC, H*C]
    const float*     Wt     = (const float*)d_in[5];      // [1, H, 3*C]
    const float*     Ws     = (const float*)d_in[6];      // [H*C, C]
    const float*     bias   = (const float*)d_in[7];      // [C]
    float*           out    = (float*)d_out;              // [N, C]

    // workspace carve-out (256B aligned)
    char* ws = (char*)d_ws;
    size_t off = 0;
    auto carve = [&](size_t bytes) -> char* {
        char* p = ws + off;
        off = (off + bytes + 255) & ~(size_t)255;
        return p;
    };
    float*    xp     = (float*)carve((size_t)N_NODES * HC * sizeof(float));   // 38.4 MB
    float*    ep     = (float*)carve((size_t)N_EDGES * HC * sizeof(float));   // 614.4 MB
    float*    alpha  = (float*)carve((size_t)N_EDGES * HEADS * sizeof(float));// 9.6 MB
    unsigned* segmax = (unsigned*)carve((size_t)N_NODES * HEADS * sizeof(unsigned));
    float*    segsum = (float*)carve((size_t)N_NODES * HEADS * sizeof(float));
    float*    aggr   = (float*)carve((size_t)N_NODES * HC * sizeof(float));   // 38.4 MB

    // 0) per-call zero init of accumulators
    {
        const int n = N_NODES * HC;
        init_state<<<(n + 255) / 256, 256, 0, stream>>>(aggr, segsum, segmax);
    }

    // 1) xp = x @ Wn : [50000,64] x [64,192]
    {
        const int tiles = (N_NODES / 16) * (HC / 16);     // 3125 * 12
        wmma_gemm_f32<CH, HC, false><<<(tiles + 7) / 8, 256, 0, stream>>>(
            x, Wn, xp, nullptr, N_NODES);
    }

    // 2) ep = edge_attr @ We : [800000,32] x [32,192]
    {
        const int tiles = (N_EDGES / 16) * (HC / 16);     // 50000 * 12
        wmma_gemm_f32<ECH, HC, false><<<(tiles + 7) / 8, 256, 0, stream>>>(
            eattr, We, ep, nullptr, N_EDGES);
    }

    // 3) attention logits + leaky relu + segment max
    {
        const int blocks = (N_EDGES + 7) / 8;             // 8 waves/block, wave/edge
        alpha_kernel<<<blocks, 256, 0, stream>>>(xp, ep, eidx, Wt, alpha, segmax);
    }

    // 4) exp + segment sum
    {
        const int n = N_EDGES * HEADS;
        softmax_expsum<<<(n + 255) / 256, 256, 0, stream>>>(alpha, segmax, eidx, segsum);
    }

    // 5) normalized message scatter-add
    {
        const int blocks = (N_EDGES + 7) / 8;
        scatter_msg<<<blocks, 256, 0, stream>>>(xp, ep, eidx, alpha, segsum, aggr);
    }

    // 6) out = aggr @ Ws + bias : [50000,192] x [192,64]
    {
        const int tiles = (N_NODES / 16) * (CH / 16);     // 3125 * 4
        wmma_gemm_f32<HC, CH, true><<<(tiles + 7) / 8, 256, 0, stream>>>(
            aggr, Ws, out, bias, N_NODES);
    }
}